// DenseEdgeEncoder_17377437679642
// MI455X (gfx1250) — compile-verified
//
#include <hip/hip_runtime.h>
#include <hip/hip_bf16.h>
#include <stdint.h>

typedef float f32x4 __attribute__((ext_vector_type(4)));

#define EMB   256   // embedding dim
#define NLOC  128   // nodes per graph == edges per graph
#define KNEI  8     // e2e neighbors per edge-node
#define NGRAPH 8

// One block per output row (tensor, g, i).  blockIdx.x in [0, 2*NGRAPH*NLOC).
//   b <  1024 : edge_dense row   b >= 1024 : e2e_dense row
// Thread t: channel chunk c4 = (t&63)*4 (float4), j-phase jb = t>>6 (uniform per wave).
__global__ __launch_bounds__(256) void dense_edge_fill_kernel(
    const float* __restrict__ x,
    const float* __restrict__ edge_attr,
    const float* __restrict__ enc_W,
    const float* __restrict__ e2e_W,
    float* __restrict__ out)
{
    __shared__ float bgRow[EMB];   // background vector W[2], staged for async LDS->mem stores

    const int rows   = NGRAPH * NLOC;          // 1024 rows per tensor
    const int b      = blockIdx.x;
    const int tensor = (b >= rows) ? 1 : 0;
    const int r      = b - tensor * rows;
    const int g      = r >> 7;                 // r / 128
    const int i      = r & 127;
    const int t      = threadIdx.x;
    const int c4     = (t & 63) << 2;          // float offset of this lane's 16B chunk
    const int jb     = t >> 6;                 // 0..3, uniform within a wave

    const float* __restrict__ W = tensor ? e2e_W : enc_W;

    // Stage background row (masked-embedding index 2 == "disconnected") into LDS.
    bgRow[t] = W[2 * EMB + t];

    // Diagonal row value (masked-embedding index 1).
    f32x4 diag = *(const f32x4*)(W + EMB + c4);

    // Special value for this row:
    //   e2e : x2[dst(i)]            = x[g*128 + (i+1)%128] + edge_attr[g*128 + i]
    //   edge: edge_attr + x_src + x_dst = x2[dst(i)] + x[g*128 + i]
    const int js = (i + 1) & 127;
    f32x4 sp = *(const f32x4*)(x        + ((size_t)(g * NLOC + js)) * EMB + c4)
             + *(const f32x4*)(edge_attr + ((size_t)(g * NLOC + i )) * EMB + c4);
    if (tensor == 0)
        sp += *(const f32x4*)(x + ((size_t)(g * NLOC + i)) * EMB + c4);

    // Output row base: rows are enumerated in exactly output order by b.
    float* __restrict__ orow = out + (size_t)b * (size_t)(NLOC * EMB);

    __syncthreads();  // bgRow visible to all waves before async stores read LDS

    // LDS byte offset of this lane's 16B chunk of bgRow (low 32 bits of flat addr
    // are the LDS offset per the LDS aperture mapping).
    const uint32_t lds_off = (uint32_t)(uintptr_t)(&bgRow[c4]);

    #pragma unroll
    for (int it = 0; it < NLOC / 4; ++it) {
        const int j = jb + (it << 2);                       // uniform per wave
        float* p = orow + (size_t)j * EMB + c4;
        bool isDiag = (j == i);
        bool isSpec = tensor ? (((j - i - 1) & 127) < KNEI) // j in {i+1..i+8} mod 128
                             : (j == js);
        if (!isDiag && !isSpec) {
            // Background fill: async DMA from LDS to memory, non-temporal,
            // never touches VGPRs. Tracked by ASYNCcnt; S_ENDPGM's implicit
            // wait-idle covers completion. Disjoint addresses from the
            // VGPR-path stores below, so no inter-counter ordering needed.
            uint64_t ga = (uint64_t)(uintptr_t)p;
            asm volatile(
                "global_store_async_from_lds_b128 %0, %1, off th:TH_STORE_NT"
                :: "v"(ga), "v"(lds_off) : "memory");
        } else {
            f32x4 v = isSpec ? sp : diag;
            __builtin_nontemporal_store(v, (f32x4*)p);      // global_store_b128 th:TH_STORE_NT
        }
    }
}

extern "C" void kernel_launch(void* const* d_in, const int* in_sizes, int n_in,
                              void* d_out, int out_size, void* d_ws, size_t ws_size,
                              hipStream_t stream) {
    const float* x         = (const float*)d_in[0];
    const float* edge_attr = (const float*)d_in[1];
    const float* enc_W     = (const float*)d_in[2];
    const float* e2e_W     = (const float*)d_in[3];
    float*       out       = (float*)d_out;

    // 2 tensors * 8 graphs * 128 rows = 2048 blocks; each block writes one
    // [128,256] fp32 row (128 KB) -> 256 MB total, write-bandwidth bound.
    dim3 grid(2 * NGRAPH * NLOC);
    dim3 block(256);
    dense_edge_fill_kernel<<<grid, block, 0, stream>>>(x, edge_attr, enc_W, e2e_W, out);
}